// VectorEncoder_16475494548009
// MI455X (gfx1250) — compile-verified
//
#include <hip/hip_runtime.h>

typedef float v2f __attribute__((ext_vector_type(2)));
typedef float v4f __attribute__((ext_vector_type(4)));
typedef float v8f __attribute__((ext_vector_type(8)));

#define N_PTS 32768
#define DIM 256
#define KCODES 4096
#define ROWS_PER_WG 64
#define LDS_STRIDE 260   // 256 + 4-float pad -> conflict-free ds_load_b64 frag reads

// LDS layout (in floats)
#define X_OFF   0
#define CB0_OFF (ROWS_PER_WG * LDS_STRIDE)            // 16640
#define CB1_OFF (CB0_OFF + 16 * LDS_STRIDE)           // 20800
#define CSQ_OFF (CB1_OFF + 16 * LDS_STRIDE)           // 24960
#define LDS_FLOATS (CSQ_OFF + KCODES)                 // 29056 -> 116,224 bytes

// ---- CDNA5 async global->LDS copy (VGLOBAL GVS form), tracked by ASYNCcnt ----
// LDS byte address from the vdst VGPR; global address = SGPR base + 32-bit VGPR byte offset
static __device__ __forceinline__ void async_cp16(unsigned long long sbase,
                                                  unsigned gbyteoff,
                                                  unsigned ldsbyteoff) {
  asm volatile("global_load_async_to_lds_b128 %0, %1, %2"
               :: "v"(ldsbyteoff), "v"(gbyteoff), "s"(sbase)
               : "memory");
}
static __device__ __forceinline__ void wait_async_le8() {
#if __has_builtin(__builtin_amdgcn_s_wait_asynccnt)
  __builtin_amdgcn_s_wait_asynccnt(8);
#else
  asm volatile("s_wait_asynccnt 0x8" ::: "memory");
#endif
}
static __device__ __forceinline__ void wait_async_le0() {
#if __has_builtin(__builtin_amdgcn_s_wait_asynccnt)
  __builtin_amdgcn_s_wait_asynccnt(0);
#else
  asm volatile("s_wait_asynccnt 0x0" ::: "memory");
#endif
}

// ---- c_sq[k] = sum_d cb[k][d]^2 ----
__global__ void vq_csq_kernel(const float* __restrict__ cb, float* __restrict__ csq) {
  int k = blockIdx.x * blockDim.x + threadIdx.x;
  if (k >= KCODES) return;
  const float4* r = reinterpret_cast<const float4*>(cb + (size_t)k * DIM);
  float s = 0.f;
#pragma unroll 4
  for (int i = 0; i < DIM / 4; ++i) {
    float4 v = r[i];
    s += v.x * v.x + v.y * v.y + v.z * v.z + v.w * v.w;
  }
  csq[k] = s;
}

// ---- main argmin via V_WMMA_F32_16X16X4_F32, async-pipelined cb staging ----
__global__ __launch_bounds__(128)
void vq_argmin_kernel(const float* __restrict__ x, const float* __restrict__ cb,
                      const float* __restrict__ csq,
                      float* __restrict__ out_idx_f, int* __restrict__ out_idx_i) {
  extern __shared__ float lds[];
  const int t = threadIdx.x;
  const int lane = t & 31;
  const int wave = t >> 5;
  const int half = lane >> 4;   // 0: lanes 0-15, 1: lanes 16-31
  const int l15  = lane & 15;
  const int rowbase = blockIdx.x * ROWS_PER_WG;
  const unsigned ldsbase = (unsigned)__builtin_amdgcn_groupstaticsize(); // dyn LDS start

  // Stage X tile: 64 rows x 256 f32 -> LDS (16B-aligned: 260*4B = 65*16B)
  const float4* x4 = reinterpret_cast<const float4*>(x);
#pragma unroll
  for (int i = 0; i < (ROWS_PER_WG * DIM / 4) / 128; ++i) {
    int lin = i * 128 + t;
    int row = lin >> 6;           // 64 float4 per row
    int c4  = lin & 63;
    float4 v = x4[(size_t)(rowbase + row) * (DIM / 4) + c4];
    *reinterpret_cast<float4*>(&lds[X_OFF + row * LDS_STRIDE + c4 * 4]) = v;
  }
  // Stage all c_sq (4096 f32) -> LDS once; kills the per-tile global load + loadcnt stall
  const float4* cq4 = reinterpret_cast<const float4*>(csq);
#pragma unroll
  for (int i = 0; i < (KCODES / 4) / 128; ++i) {
    int lin = i * 128 + t;
    float4 v = cq4[lin];
    *reinterpret_cast<float4*>(&lds[CSQ_OFF + lin * 4]) = v;
  }

  float minv[8];
  int   mini[8];
#pragma unroll
  for (int j = 0; j < 8; ++j) { minv[j] = 3.0e38f; mini[j] = 0; }

  // A-frag / B-frag per-lane LDS bases (identical addressing pattern for A and B)
  const float* arow  = &lds[X_OFF  + (wave * 16 + l15) * LDS_STRIDE + 2 * half];
  const float* brow0 = &lds[CB0_OFF + l15 * LDS_STRIDE + 2 * half];
  const float* brow1 = &lds[CB1_OFF + l15 * LDS_STRIDE + 2 * half];

  // ---- double-buffered async-to-LDS pipeline ----
  const unsigned long long cbbase = (unsigned long long)cb;
  auto issue_cb = [&](int tile, int buf) {
    unsigned cb_off = buf ? (unsigned)CB1_OFF : (unsigned)CB0_OFF;
#pragma unroll
    for (int i = 0; i < (16 * DIM / 4) / 128; ++i) {   // 8 x b128 per thread
      int lin = i * 128 + t;
      int row = lin >> 6;
      int c4  = lin & 63;
      unsigned gsrc = (unsigned)(((tile * 16 + row) * DIM + c4 * 4) * 4);      // bytes
      unsigned dst  = ldsbase + (cb_off + (unsigned)(row * LDS_STRIDE + c4 * 4)) * 4u;
      async_cp16(cbbase, gsrc, dst);
    }
  };

  issue_cb(0, 0);
  issue_cb(1, 1);

  for (int tile = 0; tile < KCODES / 16; ++tile) {
    // current tile's 8 per-wave async loads done (next tile's 8 may stay in flight)
    if (tile + 1 < KCODES / 16) wait_async_le8(); else wait_async_le0();
    __syncthreads();

    float cs = lds[CSQ_OFF + tile * 16 + l15];   // ds broadcast, pipelined with frags

    v8f acc = {};
    const float* brow = (tile & 1) ? brow1 : brow0;
#pragma unroll 8
    for (int d0 = 0; d0 < DIM; d0 += 4) {
      v2f a = *reinterpret_cast<const v2f*>(arow + d0);
      v2f b = *reinterpret_cast<const v2f*>(brow + d0);
      acc = __builtin_amdgcn_wmma_f32_16x16x4_f32(false, a, false, b,
                                                  (short)0, acc, false, false);
    }

    int code = tile * 16 + l15;
#pragma unroll
    for (int j = 0; j < 8; ++j) {
      float s = cs - 2.0f * acc[j];   // dist - ||x||^2 (row-constant term dropped)
      if (s < minv[j]) { minv[j] = s; mini[j] = code; }
    }

    __syncthreads();                              // all waves done reading this buffer
    if (tile + 2 < KCODES / 16) issue_cb(tile + 2, tile & 1);
  }

  // reduce (min,argmin) across the 16 lanes of each half-wave
#pragma unroll
  for (int j = 0; j < 8; ++j) {
    float v = minv[j];
    int   ix = mini[j];
#pragma unroll
    for (int m = 8; m >= 1; m >>= 1) {
      float ov = __shfl_xor(v, m, 32);
      int   oi = __shfl_xor(ix, m, 32);
      if (ov < v || (ov == v && oi < ix)) { v = ov; ix = oi; }
    }
    if (l15 == 0) {
      int row = rowbase + wave * 16 + j + (half ? 8 : 0);
      out_idx_f[row] = (float)ix;
      out_idx_i[row] = ix;
    }
  }
}

// ---- zero the one-hot matrix (536 MB, HBM-bound) with non-temporal stores ----
__global__ void vq_zero_kernel(v4f* __restrict__ p, long long n4) {
  long long i = (long long)blockIdx.x * blockDim.x + threadIdx.x;
  long long stride = (long long)gridDim.x * blockDim.x;
  v4f z = {0.f, 0.f, 0.f, 0.f};
  for (; i < n4; i += stride) __builtin_nontemporal_store(z, p + i);
}

// ---- scatter the ones ----
__global__ void vq_scatter_kernel(const int* __restrict__ idx, float* __restrict__ enc) {
  int n = blockIdx.x * blockDim.x + threadIdx.x;
  if (n < N_PTS) enc[(size_t)n * KCODES + idx[n]] = 1.0f;
}

extern "C" void kernel_launch(void* const* d_in, const int* in_sizes, int n_in,
                              void* d_out, int out_size, void* d_ws, size_t ws_size,
                              hipStream_t stream) {
  (void)in_sizes; (void)n_in; (void)out_size; (void)ws_size;
  const float* x  = (const float*)d_in[0];   // [N, D] fp32
  const float* cb = (const float*)d_in[1];   // [K, D] fp32

  float* out       = (float*)d_out;
  float* out_idx_f = out;                    // first N: encoding_indices (as float)
  float* enc       = out + N_PTS;            // next N*K: one-hot encodings

  float* csq   = (float*)d_ws;                                   // 4096 f32
  int*   idxws = (int*)((char*)d_ws + KCODES * sizeof(float));   // 32768 i32

  vq_csq_kernel<<<KCODES / 256, 256, 0, stream>>>(cb, csq);

  size_t shmem = (size_t)LDS_FLOATS * sizeof(float);             // 116,224 B
  vq_argmin_kernel<<<N_PTS / ROWS_PER_WG, 128, shmem, stream>>>(x, cb, csq,
                                                                out_idx_f, idxws);

  long long n4 = (long long)N_PTS * KCODES / 4;
  vq_zero_kernel<<<8192, 256, 0, stream>>>((v4f*)enc, n4);

  vq_scatter_kernel<<<N_PTS / 256, 256, 0, stream>>>(idxws, enc);
}